// AggregateGCN_19189913879214
// MI455X (gfx1250) — compile-verified
//
#include <hip/hip_runtime.h>
#include <hip/hip_bf16.h>

typedef __attribute__((ext_vector_type(2))) float v2f;
typedef __attribute__((ext_vector_type(8))) float v8f;

#define HDIM 32
#define GMAX 64
#define EMBD 16

// ---------------- utility: zero a float buffer ----------------
__global__ void zero_f32_kernel(float* __restrict__ p, long n) {
    long i = (long)blockIdx.x * blockDim.x + threadIdx.x;
    long stride = (long)gridDim.x * blockDim.x;
    for (; i < n; i += stride) p[i] = 0.0f;
}

// ---------------- degree accumulation ----------------
__global__ void degree_kernel(const int* __restrict__ src, const int* __restrict__ dst,
                              float* __restrict__ odeg, float* __restrict__ ideg, int E) {
    int i = blockIdx.x * blockDim.x + threadIdx.x;
    int stride = gridDim.x * blockDim.x;
    for (; i < E; i += stride) {
        atomicAdd(&odeg[src[i]], 1.0f);
        atomicAdd(&ideg[dst[i]], 1.0f);
    }
}

__global__ void norm_kernel(float* __restrict__ odeg, float* __restrict__ ideg, int n) {
    int i = blockIdx.x * blockDim.x + threadIdx.x;
    if (i < n) {
        odeg[i] = rsqrtf(fmaxf(odeg[i], 1.0f));
        ideg[i] = rsqrtf(fmaxf(ideg[i], 1.0f));
    }
}

// ---------------- dense transform: T = diag(out_norm) * X * W via WMMA f32 16x16x4 ----------------
// One wave computes a 16-row x 32-col output tile. W (K x 32) staged in LDS.
// A fragment (16x4 f32): lanes 0-15 -> row M=lane, K = {k0, k0+1}; lanes 16-31 -> K = {k0+2, k0+3}.
// B fragment (4x16 f32): lanes 0-15 -> col N=lane, K rows {k0, k0+1}; lanes 16-31 -> {k0+2, k0+3}.
// D (16x16 f32): VGPR r -> row r (lanes 0-15) / row r+8 (lanes 16-31), col = lane & 15.
template<int K>
__global__ void gemm_wmma_kernel(const float* __restrict__ X,        // n x K
                                 const float* __restrict__ W,        // K x 32
                                 const float* __restrict__ out_norm, // n
                                 float* __restrict__ T,              // n x 32
                                 int n) {
    __shared__ float sW[K * HDIM];
    int tid = threadIdx.x;
    for (int i = tid; i < K * HDIM; i += blockDim.x) sW[i] = W[i];
    __syncthreads();

    int wave = tid >> 5;
    int lane = tid & 31;
    int tile = blockIdx.x * 8 + wave;               // wave-uniform
    if (tile * 16 >= n) return;                     // uniform exit: EXEC stays all-1s for WMMA

    int row0 = tile * 16;
    int hi   = lane >> 4;                           // 0: lanes 0-15, 1: lanes 16-31
    int col  = lane & 15;
    int mr   = row0 + col;
    int mrc  = (mr < n) ? mr : (n - 1);
    float nrm = out_norm[mrc];

    const float* arow = X + (long)mrc * K + 2 * hi;

    v8f acc0 = {};
    v8f acc1 = {};
    for (int k0 = 0; k0 < K; k0 += 4) {
        v2f a = *(const v2f*)(arow + k0);           // 8B aligned
        a.x *= nrm; a.y *= nrm;
        int kr = k0 + 2 * hi;
        v2f b0, b1;
        b0.x = sW[kr * HDIM + col];
        b0.y = sW[(kr + 1) * HDIM + col];
        b1.x = sW[kr * HDIM + 16 + col];
        b1.y = sW[(kr + 1) * HDIM + 16 + col];
        acc0 = __builtin_amdgcn_wmma_f32_16x16x4_f32(false, a, false, b0, (short)0, acc0, false, false);
        acc1 = __builtin_amdgcn_wmma_f32_16x16x4_f32(false, a, false, b1, (short)0, acc1, false, false);
    }

    int mbase = row0 + 8 * hi;
#pragma unroll
    for (int r = 0; r < 8; ++r) {
        int m = mbase + r;
        if (m < n) {
            T[(long)m * HDIM + col]      = acc0[r];
            T[(long)m * HDIM + 16 + col] = acc1[r];
        }
    }
}

// ---------------- edge scatter-add: agg[dst] += T[src], one wave per edge ----------------
__global__ void scatter_kernel(const float* __restrict__ T, const int* __restrict__ src,
                               const int* __restrict__ dst, float* __restrict__ agg, int E) {
    int lane = threadIdx.x & 31;
    int wave = blockIdx.x * (blockDim.x >> 5) + (threadIdx.x >> 5);
    int nwaves = gridDim.x * (blockDim.x >> 5);
    for (int e = wave; e < E; e += nwaves) {
        int s = src[e];
        int d = dst[e];
        float v = T[(long)s * HDIM + lane];                 // coalesced 128B row gather (L2 hit)
        atomicAdd(&agg[(long)d * HDIM + lane], v);          // 32 consecutive f32 atomics at L2
    }
}

// ---------------- agg * in_norm + bias, ReLU (in place) ----------------
__global__ void finalize_kernel(float* __restrict__ agg, const float* __restrict__ in_norm,
                                const float* __restrict__ bias, int n) {
    long i = (long)blockIdx.x * blockDim.x + threadIdx.x;
    if (i < (long)n * HDIM) {
        int node = (int)(i >> 5);
        int f = (int)(i & 31);
        float v = agg[i] * in_norm[node] + bias[f];
        agg[i] = v > 0.0f ? v : 0.0f;
    }
}

// ---------------- per-graph pooling with LDS-local accumulation ----------------
__global__ void pool_kernel(const float* __restrict__ x, const int* __restrict__ gid,
                            float* __restrict__ gsum, float* __restrict__ gcnt, int n) {
    __shared__ float ssum[GMAX * HDIM];
    __shared__ float scnt[GMAX];
    int tid = threadIdx.x;
    for (int i = tid; i < GMAX * HDIM; i += blockDim.x) ssum[i] = 0.0f;
    for (int i = tid; i < GMAX; i += blockDim.x) scnt[i] = 0.0f;
    __syncthreads();

    long total = (long)n * HDIM;
    long i = (long)blockIdx.x * blockDim.x + tid;
    long stride = (long)gridDim.x * blockDim.x;
    for (; i < total; i += stride) {
        int node = (int)(i >> 5);
        int f = (int)(i & 31);
        int g = gid[node];
        atomicAdd(&ssum[g * HDIM + f], x[i]);
        if (f == 0) atomicAdd(&scnt[g], 1.0f);
    }
    __syncthreads();
    for (int j = tid; j < GMAX * HDIM; j += blockDim.x) atomicAdd(&gsum[j], ssum[j]);
    for (int j = tid; j < GMAX; j += blockDim.x) atomicAdd(&gcnt[j], scnt[j]);
}

// ---------------- out = (gsum / count) @ We + be ----------------
__global__ void out_kernel(const float* __restrict__ gsum, const float* __restrict__ gcnt,
                           const float* __restrict__ We, const float* __restrict__ be,
                           float* __restrict__ out) {
    int g = blockIdx.x;
    int j = threadIdx.x;
    if (j < EMBD) {
        float inv = 1.0f / fmaxf(gcnt[g], 1.0f);
        float a = be[j];
        for (int k = 0; k < HDIM; ++k)
            a += (gsum[g * HDIM + k] * inv) * We[k * EMBD + j];
        out[g * EMBD + j] = a;
    }
}

extern "C" void kernel_launch(void* const* d_in, const int* in_sizes, int n_in,
                              void* d_out, int out_size, void* d_ws, size_t ws_size,
                              hipStream_t stream) {
    (void)n_in; (void)out_size; (void)ws_size;
    const float* h   = (const float*)d_in[0];
    const int*   src = (const int*)d_in[1];
    const int*   dst = (const int*)d_in[2];
    const int*   gid = (const int*)d_in[3];
    const float* W1  = (const float*)d_in[4];
    const float* b1  = (const float*)d_in[5];
    const float* W2  = (const float*)d_in[6];
    const float* b2  = (const float*)d_in[7];
    const float* W3  = (const float*)d_in[8];
    const float* b3  = (const float*)d_in[9];
    const float* We  = (const float*)d_in[10];
    const float* be  = (const float*)d_in[11];
    float* out = (float*)d_out;

    const int N = in_sizes[3];   // graph_ids has N entries
    const int E = in_sizes[1];   // src has E entries

    // workspace layout (floats)
    float* ws   = (float*)d_ws;
    float* odeg = ws;                          // N  (becomes out_norm)
    float* ideg = odeg + N;                    // N  (becomes in_norm)
    float* gsum = ideg + N;                    // GMAX*HDIM
    float* gcnt = gsum + GMAX * HDIM;          // GMAX
    float* bufA = gcnt + GMAX;                 // N*HDIM  (transform result)
    float* bufB = bufA + (long)N * HDIM;       // N*HDIM  (aggregation / layer output)

    long hdr = 2L * N + GMAX * HDIM + GMAX;
    zero_f32_kernel<<<1024, 256, 0, stream>>>(ws, hdr);
    degree_kernel<<<2048, 256, 0, stream>>>(src, dst, odeg, ideg, E);
    norm_kernel<<<(N + 255) / 256, 256, 0, stream>>>(odeg, ideg, N);

    int tiles = (N + 15) / 16;
    int gemmBlocks = (tiles + 7) / 8;          // 8 waves (tiles) per 256-thread block
    long nh = (long)N * HDIM;
    int zb = (int)((nh + 255) / 256);

    // ---- layer 1 ----
    gemm_wmma_kernel<128><<<gemmBlocks, 256, 0, stream>>>(h, W1, odeg, bufA, N);
    zero_f32_kernel<<<4096, 256, 0, stream>>>(bufB, nh);
    scatter_kernel<<<8192, 256, 0, stream>>>(bufA, src, dst, bufB, E);
    finalize_kernel<<<zb, 256, 0, stream>>>(bufB, ideg, b1, N);

    // ---- layer 2 ----
    gemm_wmma_kernel<32><<<gemmBlocks, 256, 0, stream>>>(bufB, W2, odeg, bufA, N);
    zero_f32_kernel<<<4096, 256, 0, stream>>>(bufB, nh);
    scatter_kernel<<<8192, 256, 0, stream>>>(bufA, src, dst, bufB, E);
    finalize_kernel<<<zb, 256, 0, stream>>>(bufB, ideg, b2, N);

    // ---- layer 3 ----
    gemm_wmma_kernel<32><<<gemmBlocks, 256, 0, stream>>>(bufB, W3, odeg, bufA, N);
    zero_f32_kernel<<<4096, 256, 0, stream>>>(bufB, nh);
    scatter_kernel<<<8192, 256, 0, stream>>>(bufA, src, dst, bufB, E);
    finalize_kernel<<<zb, 256, 0, stream>>>(bufB, ideg, b3, N);

    // ---- pooling + output projection ----
    pool_kernel<<<1024, 256, 0, stream>>>(bufB, gid, gsum, gcnt, N);
    out_kernel<<<GMAX, 32, 0, stream>>>(gsum, gcnt, We, be, out);
}